// Ive_86723979640946
// MI455X (gfx1250) — compile-verified
//
#include <hip/hip_runtime.h>

// i0e(x) = exp(-|x|) * I0(x), elementwise over 64*1024*1024 fp32 values.
// Memory-bound (3.7 flops/byte vs 23.3 TB/s HBM): the only goal is saturating
// HBM with b128 non-temporal streaming loads/stores; all math is VALU filler.

typedef __attribute__((ext_vector_type(4))) float fvec4;

__device__ __forceinline__ float i0e_one(float x) {
    float ax = __builtin_fabsf(x);

    // ---- small-argument branch: exp(-ax) * P((ax/3.75)^2), A&S 9.8.1 ----
    float t = ax * 0.26666666666666666f;   // ax / 3.75
    t = t * t;
    float ps;
    ps = __builtin_fmaf(t, 0.0045813f, 0.0360768f);
    ps = __builtin_fmaf(t, ps, 0.2659732f);
    ps = __builtin_fmaf(t, ps, 1.2067492f);
    ps = __builtin_fmaf(t, ps, 3.0899424f);
    ps = __builtin_fmaf(t, ps, 3.5156229f);
    ps = __builtin_fmaf(t, ps, 1.0f);
    float small_v = __expf(-ax) * ps;      // v_exp_f32; underflows to 0 for big ax (unused there)

    // ---- large-argument branch: Q(3.75/ax) / sqrt(ax), A&S 9.8.2 ----
    float axl = fmaxf(ax, 3.75f);          // clamp so rcp/rsqrt are safe when small branch wins
    float u = __fdividef(3.75f, axl);      // v_rcp_f32 + mul (fast, ~1 ulp of need)
    float pl;
    pl = __builtin_fmaf(u, 0.00392377f, -0.01647633f);
    pl = __builtin_fmaf(u, pl, 0.02635537f);
    pl = __builtin_fmaf(u, pl, -0.02057706f);
    pl = __builtin_fmaf(u, pl, 0.00916281f);
    pl = __builtin_fmaf(u, pl, -0.00157565f);
    pl = __builtin_fmaf(u, pl, 0.00225319f);
    pl = __builtin_fmaf(u, pl, 0.01328592f);
    pl = __builtin_fmaf(u, pl, 0.39894228f);
    float large_v = pl * __frsqrt_rn(axl); // v_rsq_f32

    return (ax <= 3.75f) ? small_v : large_v;
}

__global__ __launch_bounds__(256) void i0e_kernel(const float* __restrict__ in,
                                                  float* __restrict__ out,
                                                  long long n4, long long n) {
    long long i = (long long)blockIdx.x * blockDim.x + threadIdx.x;

    if (i < n4) {
        // 128-bit non-temporal streaming load (global_load_b128, TH=NT)
        const fvec4* in4 = reinterpret_cast<const fvec4*>(in);
        fvec4 v = __builtin_nontemporal_load(in4 + i);

        fvec4 r;
        r.x = i0e_one(v.x);
        r.y = i0e_one(v.y);
        r.z = i0e_one(v.z);
        r.w = i0e_one(v.w);

        // 128-bit non-temporal streaming store (global_store_b128, TH=NT)
        fvec4* out4 = reinterpret_cast<fvec4*>(out);
        __builtin_nontemporal_store(r, out4 + i);
    }

    // Tail (n % 4 != 0): first few threads of the grid each take one scalar.
    long long base = n4 * 4;
    long long tail = n - base;           // 0 for this problem (64M % 4 == 0)
    if (i < tail) {
        out[base + i] = i0e_one(in[base + i]);
    }
}

extern "C" void kernel_launch(void* const* d_in, const int* in_sizes, int n_in,
                              void* d_out, int out_size, void* d_ws, size_t ws_size,
                              hipStream_t stream) {
    (void)n_in; (void)d_ws; (void)ws_size;

    const float* z = reinterpret_cast<const float*>(d_in[0]);
    float* out = reinterpret_cast<float*>(d_out);

    long long n  = (long long)in_sizes[0];   // 64*1024*1024 = 67,108,864
    long long n4 = n >> 2;                   // float4 count

    const int block = 256;                   // 8 wave32s per block
    long long work = (n4 > 0) ? n4 : 1;      // ensure tail threads exist even if n < 4
    unsigned int grid = (unsigned int)((work + block - 1) / block);

    i0e_kernel<<<grid, block, 0, stream>>>(z, out, n4, n);
}